// Model_88098369175826
// MI455X (gfx1250) — compile-verified
//
#include <hip/hip_runtime.h>
#include <hip/hip_bf16.h>

// ---------------------------------------------------------------------------
// CDNA5 (gfx1250, wave32) implementation.
// Attention is reassociated (no softmax in reference): av = (V K^T / H) q,
// turning O(B*T*T*H) into O(B*H*H*T). All GEMMs use V_WMMA_F32_16X16X4_F32
// (exact f32). H=10 padded to 16; padded rows/cols are materialized as zeros
// in LDS so the WMMA inner loops are branchless. The front-GEMM weight tile
// is staged into LDS with the Tensor Data Mover (tensor_load_to_lds +
// s_wait_tensorcnt). Sequential LSNN + LI classifier fused into one
// register-resident wave32 scan per batch (cross-lane __shfl matvec).
// ---------------------------------------------------------------------------

typedef __attribute__((ext_vector_type(2))) float        v2f;
typedef __attribute__((ext_vector_type(8))) float        v8f;
typedef __attribute__((ext_vector_type(4))) unsigned int v4u;
typedef __attribute__((ext_vector_type(8))) int          v8i;
typedef __attribute__((ext_vector_type(4))) int          v4i;

#define WMMA_F32(a, b, c) \
  __builtin_amdgcn_wmma_f32_16x16x4_f32(false, (a), false, (b), (short)0, (c), false, false)

#define HDIM   10
#define ODIM   2
#define BATCH  32
#define CDIM   64
#define TDIM   1024
#define NCOL   (BATCH * TDIM)          // 32768
#define PLANE  (16 * NCOL)             // 524288 floats per padded (16 x N) tensor
#define F_VTH        0.5f
#define F_A_MEM      0.1f              // DT * TAU_MEM_INV
#define F_D_SYN      0.8f              // 1 - DT * TAU_SYN_INV
#define F_D_MEM      0.9f              // 1 - DT * TAU_MEM_INV

// ---------------------------------------------------------------------------
// Front GEMM: y[16 x N] = W_ef(10x64, zero-padded to 16) * X(64 x N),
// X(c, n) = beeg[b, c, t] with n = b*T + t. One wave per 16x16 tile,
// 16 chained v_wmma_f32_16x16x4_f32 ops over K=64. Weight tile staged in
// LDS by the Tensor Data Mover (2D D#: 64x10 f32, stride 64).
// ---------------------------------------------------------------------------
__global__ void gemm_front(const float* __restrict__ W,     // 10 x 64
                           const float* __restrict__ beeg,  // B x 64 x T
                           float* __restrict__ y)           // 16 x N
{
  __shared__ float shW[16 * CDIM];                           // 4 KB, padded
  const int tid = threadIdx.x;

  // zero the padded rows 10..15 (6*64 floats)
  for (int i = tid; i < (16 - HDIM) * CDIM; i += 256)
    shW[HDIM * CDIM + i] = 0.0f;

  // wave 0: TDM DMA of the 10x64 weight tile into LDS rows 0..9
  if ((tid >> 5) == 0) {
    const unsigned int lds = (unsigned int)(uintptr_t)(&shW[0]);
    const unsigned long long ga = (unsigned long long)(uintptr_t)W;
    v4u g0; v8i g1; v4i g2 = {0, 0, 0, 0}; v4i g3 = {0, 0, 0, 0};
    v8i g4 = {0, 0, 0, 0, 0, 0, 0, 0};
    g0[0] = 1u;                                             // count=1, user D#
    g0[1] = lds;                                            // lds_addr (bytes)
    g0[2] = (unsigned int)(ga & 0xffffffffull);             // global_addr[31:0]
    g0[3] = (unsigned int)((ga >> 32) & 0x01ffffffull)      // global_addr[56:32]
          | (2u << 30);                                     // type = 2 (image)
    g1[0] = (int)(2u << 16);                                // data_size = 4B
    g1[1] = (int)((unsigned)CDIM << 16);                    // tensor_dim0 = 64
    g1[2] = (int)((unsigned)HDIM << 16);                    // tensor_dim1 = 10
    g1[3] = (int)((unsigned)CDIM << 16);                    // tile_dim0 = 64
    g1[4] = (int)HDIM;                                      // tile_dim1 = 10
    g1[5] = (int)CDIM;                                      // dim0_stride = 64
    g1[6] = 0; g1[7] = 0;
    __builtin_amdgcn_tensor_load_to_lds(g0, g1, g2, g3, g4, 0);
    __builtin_amdgcn_s_wait_tensorcnt(0);
  }
  __syncthreads();

  const int lane  = tid & 31;
  const int wave  = tid >> 5;
  const int tile  = blockIdx.x * (blockDim.x >> 5) + wave;   // 0..2047
  const int nbase = tile << 4;
  const int b     = nbase >> 10;
  const int t     = (nbase & 1023) + (lane & 15);
  const int m     = lane & 15;
  const int khalf = (lane >> 4) << 1;                        // 0 or 2

  const v2f*   w2    = (const v2f*)shW;
  const int    wbase = (m * CDIM + khalf) >> 1;              // v2f index
  const float* bbase = beeg + (size_t)b * CDIM * TDIM + (size_t)khalf * TDIM + t;
  __builtin_prefetch(bbase, 0, 1);                           // global_prefetch_b8

  v8f acc = {};
#pragma unroll
  for (int k0 = 0; k0 < CDIM; k0 += 4) {
    v2f bm;
    bm.x = bbase[(size_t)k0 * TDIM];                         // immediate offsets
    bm.y = bbase[(size_t)(k0 + 1) * TDIM];
    acc = WMMA_F32(w2[wbase + (k0 >> 1)], bm, acc);
  }
#pragma unroll
  for (int i = 0; i < 8; ++i) {
    const int row = i + ((lane >> 4) << 3);
    y[(size_t)row * NCOL + nbase + (lane & 15)] = acc[i];
  }
}

// ---------------------------------------------------------------------------
// Generic 16xN GEMM with K=16: y = A(16x16) * X(16 x N).
// MODE 0: A = global weight, 10x10 row-major, staged zero-padded in LDS.
// MODE 1: A = per-batch 16x16 matrix Mmat[b] (b derived from the column tile).
// TRANS 1: store D as y[n*16 + m] (feeds the time scan).
// ---------------------------------------------------------------------------
template <int MODE, int TRANS>
__global__ void gemm_h16(const float* __restrict__ A,
                         const float* __restrict__ X,   // 16 x N
                         float* __restrict__ y)
{
  __shared__ float shW[256];
  const int tid = threadIdx.x;
  if (MODE == 0) {
    const int row = tid >> 4, cw = tid & 15;                 // one pass, 256 thr
    shW[tid] = (row < HDIM && cw < HDIM) ? A[row * HDIM + cw] : 0.0f;
    __syncthreads();
  }

  const int lane  = tid & 31;
  const int wave  = tid >> 5;
  const int tile  = blockIdx.x * (blockDim.x >> 5) + wave;
  const int nbase = tile << 4;
  const int m     = lane & 15;
  const int col   = lane & 15;
  const int khalf = (lane >> 4) << 1;

  v2f afrag[4];
  if (MODE == 0) {
    const v2f* w2 = (const v2f*)shW;
#pragma unroll
    for (int j = 0; j < 4; ++j) afrag[j] = w2[(m * 16 + khalf + 4 * j) >> 1];
  } else {
    const int b = nbase >> 10;
    const float* Ab = A + b * 256 + m * 16 + khalf;
#pragma unroll
    for (int j = 0; j < 4; ++j) { afrag[j].x = Ab[4 * j]; afrag[j].y = Ab[4 * j + 1]; }
  }

  const float* xb = X + (size_t)khalf * NCOL + nbase + col;
  v8f acc = {};
#pragma unroll
  for (int j = 0; j < 4; ++j) {
    v2f bm;
    bm.x = xb[(size_t)(4 * j) * NCOL];
    bm.y = xb[(size_t)(4 * j + 1) * NCOL];
    acc = WMMA_F32(afrag[j], bm, acc);
  }
#pragma unroll
  for (int i = 0; i < 8; ++i) {
    const int row = i + ((lane >> 4) << 3);
    if (TRANS)
      y[(size_t)(nbase + col) * 16 + row] = acc[i];
    else
      y[(size_t)row * NCOL + nbase + col] = acc[i];
  }
}

// ---------------------------------------------------------------------------
// Linear-attention contraction: Mmat[b] = (V_b K_b^T) / H, 16x16 per batch,
// K-dimension = T = 1024 (256 wmma steps). One wave per batch.
// ---------------------------------------------------------------------------
__global__ void attn_vkt(const float* __restrict__ Vb,  // 16 x N
                         const float* __restrict__ Kb,  // 16 x N
                         float* __restrict__ Mmat)      // B x 16 x 16
{
  const int b     = blockIdx.x;
  const int lane  = threadIdx.x;
  const int m     = lane & 15;
  const int khalf = (lane >> 4) << 1;
  const float* vbase = Vb + (size_t)b * TDIM + (size_t)m * NCOL + khalf;
  const float* kbase = Kb + (size_t)b * TDIM + (size_t)(lane & 15) * NCOL + khalf;

  v8f acc = {};
#pragma unroll 4
  for (int t0 = 0; t0 < TDIM; t0 += 4) {
    v2f a, bm;
    a.x  = vbase[t0];
    a.y  = vbase[t0 + 1];
    bm.x = kbase[t0];
    bm.y = kbase[t0 + 1];
    acc = WMMA_F32(a, bm, acc);
  }
#pragma unroll
  for (int i = 0; i < 8; ++i) {
    const int row = i + ((lane >> 4) << 3);
    Mmat[b * 256 + row * 16 + (lane & 15)] = acc[i] * (1.0f / (float)HDIM);
  }
}

// ---------------------------------------------------------------------------
// Per-channel batch statistics: sum and sum-of-squares over the N=32768
// (B,T) samples of one channel. One 256-thread block per channel -> no atomics.
// ---------------------------------------------------------------------------
__global__ void chan_stats(const float* __restrict__ y, int ystride,
                           float* __restrict__ stats)   // [which][16][2]
{
  const int h = blockIdx.x, which = blockIdx.y, tid = threadIdx.x;
  const float* p = y + (size_t)which * ystride + (size_t)h * NCOL;
  float s1 = 0.0f, s2 = 0.0f;
  for (int n = tid; n < NCOL; n += 256) { float v = p[n]; s1 += v; s2 += v * v; }
  __shared__ float sh1[256], sh2[256];
  sh1[tid] = s1; sh2[tid] = s2; __syncthreads();
  for (int s = 128; s > 0; s >>= 1) {
    if (tid < s) { sh1[tid] += sh1[tid + s]; sh2[tid] += sh2[tid + s]; }
    __syncthreads();
  }
  if (tid == 0) {
    stats[which * 32 + h * 2 + 0] = sh1[0];
    stats[which * 32 + h * 2 + 1] = sh2[0];
  }
}

// ---------------------------------------------------------------------------
// BN (batch stats, biased var, eps=1e-5) + activation. Padded rows -> 0.
// act: 0 = tanh, 1 = relu.
// ---------------------------------------------------------------------------
__global__ void bn_act(const float* __restrict__ y, const float* __restrict__ stats,
                       const float* __restrict__ g, const float* __restrict__ be,
                       float* __restrict__ x, int act)
{
  const int idx = blockIdx.x * 256 + threadIdx.x;     // 0 .. PLANE-1
  const int h = idx >> 15;
  float out = 0.0f;
  if (h < HDIM) {
    const float inv_n = 1.0f / (float)NCOL;
    const float mean  = stats[h * 2 + 0] * inv_n;
    const float var   = stats[h * 2 + 1] * inv_n - mean * mean;
    const float r     = rsqrtf(var + 1e-5f);
    const float val   = g[h] * (y[idx] - mean) * r + be[h];
    out = act ? fmaxf(val, 0.0f) : tanhf(val);
  }
  x[idx] = out;
}

// ---------------------------------------------------------------------------
// Fused LSNN (beta=0, b=VTH) + LI classifier scan + time-mean.
// One wave32 per batch; lane h holds the h-th neuron state; the 10x10
// recurrent matvec and the 2x10 classifier use cross-lane __shfl broadcasts.
// u layout: [B*T][16]; lanes 16..31 mirror lanes 0..15 (branchless) but
// their spikes are forced to zero, so they never influence the result.
// ---------------------------------------------------------------------------
__global__ void lsnn_scan(const float* __restrict__ u,
                          const float* __restrict__ W_rec,  // 10 x 10
                          const float* __restrict__ W_cls,  // 2 x 10
                          const float* __restrict__ b_cls,  // 2
                          float* __restrict__ out)          // B x 2
{
  const int b = blockIdx.x;
  const int lane = threadIdx.x;

  float wrec[HDIM], wcls[HDIM];
#pragma unroll
  for (int j = 0; j < HDIM; ++j) {
    wrec[j] = (lane < HDIM) ? W_rec[lane * HDIM + j] : 0.0f;
    wcls[j] = (lane < ODIM) ? W_cls[lane * HDIM + j] : 0.0f;
  }
  const float bcls = (lane < ODIM) ? b_cls[lane] : 0.0f;

  float z = 0.0f, vm = 0.0f, cur = 0.0f;
  float li_v = 0.0f, li_i = 0.0f, acc = 0.0f;
  const float* ub = u + (size_t)b * TDIM * 16 + (lane & 15);

  for (int t = 0; t < TDIM; ++t) {
    const float ut = ub[t * 16];                       // branchless, clamped lane
    float rec = 0.0f;
#pragma unroll
    for (int j = 0; j < HDIM; ++j)
      rec += wrec[j] * __shfl(z, j, 32);
    const float ijump = cur + ut + rec;
    const float vdec  = vm + F_A_MEM * (ijump - vm);
    cur = ijump * F_D_SYN;
    const float zn = (lane < HDIM && vdec > F_VTH) ? 1.0f : 0.0f;
    vm = (1.0f - zn) * vdec;
    z  = zn;
    float cls = 0.0f;
#pragma unroll
    for (int j = 0; j < HDIM; ++j)
      cls += wcls[j] * __shfl(z, j, 32);
    li_i = li_i * F_D_SYN + cls + bcls;
    li_v = li_v * F_D_MEM + F_A_MEM * li_i;
    acc += li_v;
  }
  if (lane < ODIM) out[b * ODIM + lane] = acc * (1.0f / (float)TDIM);
}

// ---------------------------------------------------------------------------
extern "C" void kernel_launch(void* const* d_in, const int* in_sizes, int n_in,
                              void* d_out, int out_size, void* d_ws, size_t ws_size,
                              hipStream_t stream) {
  (void)in_sizes; (void)n_in; (void)out_size; (void)ws_size;
  const float* beeg  = (const float*)d_in[2];
  const float* W_ef  = (const float*)d_in[4];
  // b_ef (d_in[5]) cancels exactly inside BN -> folded away.
  const float* g_i   = (const float*)d_in[6];
  const float* be_i  = (const float*)d_in[7];
  const float* Wq    = (const float*)d_in[8];
  const float* g_q   = (const float*)d_in[9];
  const float* be_q  = (const float*)d_in[10];
  const float* Wk    = (const float*)d_in[11];
  const float* g_k   = (const float*)d_in[12];
  const float* be_k  = (const float*)d_in[13];
  const float* Wv    = (const float*)d_in[14];
  const float* g_v   = (const float*)d_in[15];
  const float* be_v  = (const float*)d_in[16];
  const float* Wc    = (const float*)d_in[17];
  const float* g_c   = (const float*)d_in[18];
  const float* be_c  = (const float*)d_in[19];
  const float* W_in  = (const float*)d_in[20];
  const float* W_rec = (const float*)d_in[21];
  const float* W_cls = (const float*)d_in[22];
  const float* b_cls = (const float*)d_in[23];
  float* outp = (float*)d_out;

  float* ws   = (float*)d_ws;
  float* bufA = ws;                         // y0, later av
  float* bufX = ws + (size_t)PLANE;         // x
  float* bufY = ws + (size_t)2 * PLANE;     // yq|yk|yv, later yc|x2|u
  float* bufQ = ws + (size_t)5 * PLANE;
  float* bufK = ws + (size_t)6 * PLANE;
  float* bufV = ws + (size_t)7 * PLANE;
  float* Mm   = ws + (size_t)8 * PLANE;               // 32*256
  float* st   = ws + (size_t)8 * PLANE + 8192;        // stats: 5 stages * 32

  const dim3 gBlk(256);                 // 8 waves -> 8 tiles per block
  const dim3 gGrd(NCOL / 16 / 8);       // 256 blocks = 2048 tiles
  const dim3 eGrd(PLANE / 256);         // elementwise

  // 1) front embedding GEMM (f32 WMMA + TDM weight stage), BN stats + tanh
  gemm_front<<<gGrd, gBlk, 0, stream>>>(W_ef, beeg, bufA);
  chan_stats<<<dim3(HDIM, 1), 256, 0, stream>>>(bufA, 0, st + 0);
  bn_act<<<eGrd, 256, 0, stream>>>(bufA, st + 0, g_i, be_i, bufX, 0);

  // 2) q / k / v pre-activations (WMMA), stats, BN+act
  gemm_h16<0, 0><<<gGrd, gBlk, 0, stream>>>(Wq, bufX, bufY);
  gemm_h16<0, 0><<<gGrd, gBlk, 0, stream>>>(Wk, bufX, bufY + PLANE);
  gemm_h16<0, 0><<<gGrd, gBlk, 0, stream>>>(Wv, bufX, bufY + 2 * PLANE);
  chan_stats<<<dim3(HDIM, 3), 256, 0, stream>>>(bufY, PLANE, st + 32);
  bn_act<<<eGrd, 256, 0, stream>>>(bufY,             st + 32, g_q, be_q, bufQ, 1);
  bn_act<<<eGrd, 256, 0, stream>>>(bufY + PLANE,     st + 64, g_k, be_k, bufK, 1);
  bn_act<<<eGrd, 256, 0, stream>>>(bufY + 2 * PLANE, st + 96, g_v, be_v, bufV, 0);

  // 3) linear attention, reassociated: M = V K^T / H ; av = M q
  attn_vkt<<<BATCH, 32, 0, stream>>>(bufV, bufK, Mm);
  gemm_h16<1, 0><<<gGrd, gBlk, 0, stream>>>(Mm, bufQ, bufA);   // av

  // 4) output projection + BN + relu
  gemm_h16<0, 0><<<gGrd, gBlk, 0, stream>>>(Wc, bufA, bufY);   // yc
  chan_stats<<<dim3(HDIM, 1), 256, 0, stream>>>(bufY, 0, st + 128);
  bn_act<<<eGrd, 256, 0, stream>>>(bufY, st + 128, g_c, be_c, bufY + PLANE, 1); // x2

  // 5) precompute LSNN input currents u = W_in * x2, stored [n][16]
  gemm_h16<0, 1><<<gGrd, gBlk, 0, stream>>>(W_in, bufY + PLANE, bufY + 2 * PLANE);

  // 6) fused sequential LSNN + LI classifier scan (one wave per batch)
  lsnn_scan<<<BATCH, 32, 0, stream>>>(bufY + 2 * PLANE, W_rec, W_cls, b_cls, outp);
}